// Net_58213986729993
// MI455X (gfx1250) — compile-verified
//
#include <hip/hip_runtime.h>
#include <hip/hip_bf16.h>

typedef float v2f __attribute__((ext_vector_type(2)));
typedef float v8f __attribute__((ext_vector_type(8)));

#define HID    1088
#define NBLK   34
#define TSTEPS 64
#define BATCH  512
#define NFEAT  32

// Native gfx1250 V_TANH_F32 if the builtin exists; branch-free exp fallback otherwise.
__device__ __forceinline__ float fast_tanh(float x) {
#if __has_builtin(__builtin_amdgcn_tanhf)
    return __builtin_amdgcn_tanhf(x);
#elif __has_builtin(__builtin_amdgcn_tanh_f32)
    return __builtin_amdgcn_tanh_f32(x);
#else
    // tanh(x) = 1 - 2/(exp(2x)+1); correct limits at +/-inf, no branches
    return 1.0f - 2.0f * __builtin_amdgcn_rcpf(__expf(2.0f * x) + 1.0f);
#endif
}

__device__ __forceinline__ float fast_sigmoid(float x) {
    // sigmoid(x) == 0.5*tanh(x/2) + 0.5  (exact identity)
    return __builtin_fmaf(0.5f, fast_tanh(0.5f * x), 0.5f);
}

// Compile-time + scheduling fence for wave-private LDS regions.
// HW executes same-wave DS ops in order; we only need to stop compiler
// reordering, so this costs ~0 cycles (no s_barrier needed).
__device__ __forceinline__ void wave_lds_fence() {
    __builtin_amdgcn_fence(__ATOMIC_ACQ_REL, "wavefront");
    __builtin_amdgcn_wave_barrier();
}

// One full gate: D(16x16) = [x|1] @ [u,b] + h(16x32) @ V(32x16), 9 chained WMMAs.
__device__ __forceinline__ v8f gate_mm(v2f aX, v2f bX, const v2f* aF, const v2f* bF) {
    v8f acc = {0.f,0.f,0.f,0.f,0.f,0.f,0.f,0.f};
    acc = __builtin_amdgcn_wmma_f32_16x16x4_f32(false, aX, false, bX, (short)0, acc, false, false);
#pragma unroll
    for (int k = 0; k < 8; ++k)
        acc = __builtin_amdgcn_wmma_f32_16x16x4_f32(false, aF[k], false, bF[k], (short)0, acc, false, false);
    return acc;
}

__global__ __launch_bounds__(128) void lstm_block_kernel(
    const float* __restrict__ x_seq,
    const float* __restrict__ U_i, const float* __restrict__ b_i,
    const float* __restrict__ U_f, const float* __restrict__ b_f,
    const float* __restrict__ U_c, const float* __restrict__ b_c,
    const float* __restrict__ U_o, const float* __restrict__ b_o,
    const float* __restrict__ V_i, const float* __restrict__ V_f,
    const float* __restrict__ V_c, const float* __restrict__ V_o,
    const float* __restrict__ out_coef,
    float* __restrict__ ws_part)
{
    __shared__ float lds[10240];        // 40 KB: h tiles (2048) + interleaved x (8192)
    const int blk  = blockIdx.x % NBLK; // hidden block 0..33
    const int tile = blockIdx.x / NBLK; // batch tile of 64 rows
    const int wave = threadIdx.x >> 5;
    const int lane = threadIdx.x & 31;
    const int hi   = lane >> 4;         // lane half
    const int ln   = lane & 15;
    const int rowBase = tile * 64 + wave * 16;

    float* ldsH = lds + wave * 512;             // [16 rows][32 cols]   (wave-private)
    float* ldsX = lds + 2048 + wave * 2048;     // [64 t][16 m][2]      (wave-private)

    // ---- stage per-row input features (xa, xc interleaved -> one b64/lane/step) ----
    const int featA = (blk < NFEAT) ? blk : 0;  // blocks >=32 have zero coeff anyway
    for (int idx = lane; idx < TSTEPS * 16; idx += 32) {
        int m = idx & 15, t = idx >> 4;
        const float* xr = x_seq + (size_t)(rowBase + m) * (TSTEPS * NFEAT) + (size_t)t * NFEAT;
        ldsX[t * 32 + m * 2]     = xr[featA];
        ldsX[t * 32 + m * 2 + 1] = xr[1];       // feature 1 (only block 0 uses it)
    }

    // ---- rank-2 input + bias folded into one extra WMMA K-chunk per gate ----
    // B-extra layout: K rows = (ua, uc, bias, 0):
    //   lanes 0-15  (r=0,1): (ua[N], uc[N]) ; lanes 16-31: (bias[N], 0)
    v2f BX[4][2];
    {
        const float* Up[4] = {U_i, U_f, U_c, U_o};
        const float* bp[4] = {b_i, b_f, b_c, b_o};
#pragma unroll
        for (int g = 0; g < 4; ++g)
#pragma unroll
            for (int n = 0; n < 2; ++n) {
                int col = blk * 32 + n * 16 + ln;
                float ua = 0.f, uc = 0.f;
                if (blk < NFEAT) ua = Up[g][(size_t)blk * HID + col];
                if (blk == 0) {
                    ua += Up[g][(size_t)32 * HID + col];   // FEAT_SEQ[32] = feature 0
                    uc  = Up[g][(size_t)33 * HID + col];   // FEAT_SEQ[33] = feature 1
                }
                float bbv = bp[g][col];
                BX[g][n].x = hi ? bbv : ua;
                BX[g][n].y = hi ? 0.f : uc;
            }
    }

    // ---- V diagonal block -> register-resident B fragments ----
    // B layout (4x16): lane l, vgpr r -> B[K = 4k + r + 2*(l>=16), N = n*16 + l%16]
    v2f VI[2][8], VF[2][8], VG[2][8], VO[2][8];
#pragma unroll
    for (int n = 0; n < 2; ++n)
#pragma unroll
        for (int k = 0; k < 8; ++k) {
            int K0  = 4 * k + 2 * hi;
            int col = blk * 32 + n * 16 + ln;
            size_t rbase = (size_t)(blk * 32 + K0) * HID + col;
            VI[n][k].x = V_i[rbase]; VI[n][k].y = V_i[rbase + HID];
            VF[n][k].x = V_f[rbase]; VF[n][k].y = V_f[rbase + HID];
            VG[n][k].x = V_c[rbase]; VG[n][k].y = V_c[rbase + HID];
            VO[n][k].x = V_o[rbase]; VO[n][k].y = V_o[rbase + HID];
        }

    __syncthreads();   // one-time: staging complete before the recurrence

    // h A-fragments (16x4 per chunk): lane l, vgpr r -> h[M = l%16, K = 4k + r + 2*(l>=16)]
    v2f aF[8];
#pragma unroll
    for (int k = 0; k < 8; ++k) { aF[k].x = 0.f; aF[k].y = 0.f; }

    v8f c0 = {0.f,0.f,0.f,0.f,0.f,0.f,0.f,0.f};  // cell state, C/D layout, N-tile 0
    v8f c1 = c0;                                 // N-tile 1

    for (int t = 0; t < TSTEPS; ++t) {
        // A-extra chunk: lanes 0-15 carry (xa[M], xc[M]); lanes 16-31 carry (1, 0)
        v2f xv = *(const v2f*)&ldsX[t * 32 + ln * 2];   // ds_load_b64
        v2f aX;
        aX.x = hi ? 1.0f : xv.x;
        aX.y = hi ? 0.0f : xv.y;

        // ---- forget gate: c *= sigmoid([x|1]@[u,b] + h@Vf) ----
        v8f acc0 = gate_mm(aX, BX[1][0], aF, VF[0]);
        v8f acc1 = gate_mm(aX, BX[1][1], aF, VF[1]);
#pragma unroll
        for (int r = 0; r < 8; ++r) { c0[r] *= fast_sigmoid(acc0[r]); c1[r] *= fast_sigmoid(acc1[r]); }

        // ---- input gate ----
        v8f iv0, iv1;
        acc0 = gate_mm(aX, BX[0][0], aF, VI[0]);
        acc1 = gate_mm(aX, BX[0][1], aF, VI[1]);
#pragma unroll
        for (int r = 0; r < 8; ++r) { iv0[r] = fast_sigmoid(acc0[r]); iv1[r] = fast_sigmoid(acc1[r]); }

        // ---- cell gate: c += i * tanh(...) ----
        acc0 = gate_mm(aX, BX[2][0], aF, VG[0]);
        acc1 = gate_mm(aX, BX[2][1], aF, VG[1]);
#pragma unroll
        for (int r = 0; r < 8; ++r) { c0[r] += iv0[r]*fast_tanh(acc0[r]); c1[r] += iv1[r]*fast_tanh(acc1[r]); }

        // ---- output gate -> h ----
        acc0 = gate_mm(aX, BX[3][0], aF, VO[0]);
        acc1 = gate_mm(aX, BX[3][1], aF, VO[1]);

        // ldsH is wave-private: same-wave DS ops execute in order; only stop
        // compiler reordering (prev-iteration aF loads precede these stores).
        wave_lds_fence();
        // C/D layout store: lane l, vgpr r -> (M = r + 8*hi, col = n*16 + ln)
#pragma unroll
        for (int r = 0; r < 8; ++r) {
            float h0 = fast_sigmoid(acc0[r]) * fast_tanh(c0[r]);
            float h1 = fast_sigmoid(acc1[r]) * fast_tanh(c1[r]);
            int row = r + 8 * hi;
            ldsH[row * 32 + ln]      = h0;
            ldsH[row * 32 + 16 + ln] = h1;
        }
        wave_lds_fence();
        // reload as A fragments for next step (contiguous K pair -> ds_load_b64)
#pragma unroll
        for (int k = 0; k < 8; ++k) {
            int K0 = 4 * k + 2 * hi;
            aF[k].x = ldsH[ln * 32 + K0];
            aF[k].y = ldsH[ln * 32 + K0 + 1];
        }
    }

    wave_lds_fence();
    // per-block partial of h_T . out_coef (write-once -> deterministic, no atomics)
    if (lane < 16) {
        float s = 0.f;
#pragma unroll
        for (int j = 0; j < 32; ++j)
            s += ldsH[lane * 32 + j] * out_coef[blk * 32 + j];
        ws_part[blk * BATCH + rowBase + lane] = s;
    }
}

__global__ __launch_bounds__(256) void finalize_kernel(
    const float* __restrict__ ws_part,
    const float* __restrict__ x_stat,
    const float* __restrict__ W1, const float* __restrict__ b1,
    const float* __restrict__ W2, const float* __restrict__ b2,
    const float* __restrict__ out_coef, const float* __restrict__ out_bias,
    float* __restrict__ out)
{
    int row = blockIdx.x * blockDim.x + threadIdx.x;
    if (row >= BATCH) return;
    float s = out_bias[0];
#pragma unroll
    for (int b = 0; b < NBLK; ++b) s += ws_part[b * BATCH + row];
    // x_stat MLP: hid = relu(x*W1+b1); m = sigmoid(sum_k hid*W2 + b2)
#pragma unroll
    for (int f = 0; f < 16; ++f) {
        float xv  = x_stat[row * 16 + f];
        float acc = b2[f];
#pragma unroll
        for (int k = 0; k < 10; ++k) {
            float h = xv * W1[f * 10 + k] + b1[f * 10 + k];
            h = h > 0.f ? h : 0.f;
            acc += h * W2[f * 10 + k];
        }
        s += fast_sigmoid(acc) * out_coef[HID + f];
    }
    out[row] = fast_sigmoid(s);
}

extern "C" void kernel_launch(void* const* d_in, const int* in_sizes, int n_in,
                              void* d_out, int out_size, void* d_ws, size_t ws_size,
                              hipStream_t stream) {
    const float* x_seq    = (const float*)d_in[0];
    const float* x_stat   = (const float*)d_in[1];
    const float* U_i      = (const float*)d_in[2];
    const float* V_i      = (const float*)d_in[3];
    const float* b_i      = (const float*)d_in[4];
    const float* U_f      = (const float*)d_in[5];
    const float* V_f      = (const float*)d_in[6];
    const float* b_f      = (const float*)d_in[7];
    const float* U_c      = (const float*)d_in[8];
    const float* V_c      = (const float*)d_in[9];
    const float* b_c      = (const float*)d_in[10];
    const float* U_o      = (const float*)d_in[11];
    const float* V_o      = (const float*)d_in[12];
    const float* b_o      = (const float*)d_in[13];
    const float* W1       = (const float*)d_in[14];
    const float* b1       = (const float*)d_in[15];
    const float* W2       = (const float*)d_in[16];
    const float* b2       = (const float*)d_in[17];
    const float* out_coef = (const float*)d_in[18];
    const float* out_bias = (const float*)d_in[19];
    float* out     = (float*)d_out;
    float* ws_part = (float*)d_ws;   // 34*512 floats, each slot written exactly once

    dim3 grid(NBLK * (BATCH / 64));  // 272 workgroups, 128 threads (4 waves) each
    lstm_block_kernel<<<grid, 128, 0, stream>>>(
        x_seq, U_i, b_i, U_f, b_f, U_c, b_c, U_o, b_o,
        V_i, V_f, V_c, V_o, out_coef, ws_part);

    finalize_kernel<<<dim3((BATCH + 255) / 256), 256, 0, stream>>>(
        ws_part, x_stat, W1, b1, W2, b2, out_coef, out_bias, out);
}